// GraphLayerNorm_23252952940944
// MI455X (gfx1250) — compile-verified
//
#include <hip/hip_runtime.h>

// GraphNorm for MI455X (gfx1250, wave32).
// Memory-bound: ~768MB total traffic => ~33us roofline at 23.3 TB/s.
// Stats reduction uses V_WMMA_F32_16X16X4_F32 with A=ones so the matrix unit
// performs the per-channel column sums (sum and sum-of-squares chains).

#define NUM_GRAPHS 512
#define CCH 128
#define EPS 1e-5f

typedef float v2f __attribute__((ext_vector_type(2)));
typedef float v8f __attribute__((ext_vector_type(8)));

// ---------------- pass 0: zero the 512 per-graph counters ----------------
__global__ void gn_zero_counts(int* __restrict__ counts) {
    counts[threadIdx.x] = 0;
}

// ---------------- pass 1: per-graph node counts (histogram) ----------------
__global__ void gn_count(const int* __restrict__ batch, int* __restrict__ counts, int N) {
    int i = blockIdx.x * blockDim.x + threadIdx.x;
    if (i < N) atomicAdd(&counts[batch[i]], 1);
}

// ---------------- pass 2: exclusive scan -> per-graph start offsets ----------------
__global__ void gn_scan(const int* __restrict__ counts, int* __restrict__ starts) {
    __shared__ int buf[NUM_GRAPHS];
    int t = threadIdx.x;
    buf[t] = counts[t];
    __syncthreads();
    for (int off = 1; off < NUM_GRAPHS; off <<= 1) {
        int v = (t >= off) ? buf[t - off] : 0;
        __syncthreads();
        buf[t] += v;
        __syncthreads();
    }
    starts[t] = buf[t] - counts[t];   // exclusive prefix
}

// ---------------- pass 3: per-graph mean / inv_std via WMMA column-sum ----------------
// One block (8 waves) per graph. Wave w owns channels [16w, 16w+16).
// Each iteration consumes a 4-node x 16-channel tile as the WMMA B matrix
// (4x16 f32 -> 2 VGPRs). A = ones(16x4) => D rows are per-channel sums over K.
// Main loop is guard-free (uniform trip count, unconditional loads) so loads
// pipeline; only the final partial tile is predicated.
__global__ void gn_stats(const float* __restrict__ x,
                         const int* __restrict__ counts,
                         const int* __restrict__ starts,
                         float* __restrict__ mean,
                         float* __restrict__ invstd) {
    const int g     = blockIdx.x;
    const int n     = counts[g];
    const int start = starts[g];

    const int wave  = threadIdx.x >> 5;   // 0..7
    const int lane  = threadIdx.x & 31;
    const int cbase = wave * 16;
    const int grp   = lane >> 4;          // K rows {0,1} vs {2,3}
    const int col   = lane & 15;          // channel within the slab (N index)

    // lane's base pointer: row (start + 2*grp), channel cbase+col
    const float* xp = x + ((size_t)start + 2 * grp) * CCH + cbase + col;

    v2f a;  a.x = 1.0f; a.y = 1.0f;       // A = ones(16x4), identical in all lanes
    v8f accS = {};                         // per-channel sums
    v8f accQ = {};                         // per-channel sums of squares

    const int nfull = n & ~3;              // full 4-node tiles
    int i = 0;
#pragma unroll 2
    for (; i < nfull; i += 4) {
        __builtin_prefetch(xp + (size_t)(i + 24) * CCH, 0, 3);
        const float b0 = xp[(size_t)i * CCH];          // node i + 2*grp
        const float b1 = xp[(size_t)(i + 1) * CCH];    // node i + 2*grp + 1
        v2f b;  b.x  = b0;      b.y  = b1;
        v2f bq; bq.x = b0 * b0; bq.y = b1 * b1;
        accS = __builtin_amdgcn_wmma_f32_16x16x4_f32(
                   false, a, false, b,  (short)0, accS, false, false);
        accQ = __builtin_amdgcn_wmma_f32_16x16x4_f32(
                   false, a, false, bq, (short)0, accQ, false, false);
    }
    if (i < n) {                           // single predicated tail tile
        const int i0 = i + 2 * grp;
        const int i1 = i0 + 1;
        const float b0 = (i0 < n) ? xp[(size_t)i * CCH] : 0.0f;
        const float b1 = (i1 < n) ? xp[(size_t)(i + 1) * CCH] : 0.0f;
        v2f b;  b.x  = b0;      b.y  = b1;
        v2f bq; bq.x = b0 * b0; bq.y = b1 * b1;
        accS = __builtin_amdgcn_wmma_f32_16x16x4_f32(
                   false, a, false, b,  (short)0, accS, false, false);
        accQ = __builtin_amdgcn_wmma_f32_16x16x4_f32(
                   false, a, false, bq, (short)0, accQ, false, false);
    }

    // D layout: VGPR0, lanes 0..15 hold row M=0 (= column sums), N = lane.
    if (lane < 16) {
        const float cnt = (float)(n > 0 ? n : 1);
        const float rc  = 1.0f / cnt;
        const float m   = accS[0] * rc;
        const float var = accQ[0] * rc - m * m;
        const float is  = rsqrtf(var + EPS);
        mean  [g * CCH + cbase + lane] = m;
        invstd[g * CCH + cbase + lane] = is;
    }
}

// ---------------- pass 4: normalize + affine (float4 / 128-bit traffic) ----------------
__global__ void gn_norm(const float* __restrict__ x,
                        const int* __restrict__ batch,
                        const float* __restrict__ weight,
                        const float* __restrict__ bias,
                        const float* __restrict__ mean,
                        const float* __restrict__ invstd,
                        float* __restrict__ out,
                        int total4) {
    int tid = blockIdx.x * blockDim.x + threadIdx.x;   // over N * C/4
    if (tid >= total4) return;
    const int node = tid >> 5;            // C/4 = 32 float4 per node
    const int c4   = (tid & 31) << 2;     // channel base
    const int g    = batch[node];

    const float4 xv = *(const float4*)(x      + (size_t)node * CCH + c4);
    const float4 mv = *(const float4*)(mean   + g * CCH + c4);
    const float4 iv = *(const float4*)(invstd + g * CCH + c4);
    const float4 wv = *(const float4*)(weight + c4);
    const float4 bv = *(const float4*)(bias   + c4);

    float4 o;
    o.x = (xv.x - mv.x) * iv.x * wv.x + bv.x;
    o.y = (xv.y - mv.y) * iv.y * wv.y + bv.y;
    o.z = (xv.z - mv.z) * iv.z * wv.z + bv.z;
    o.w = (xv.w - mv.w) * iv.w * wv.w + bv.w;

    *(float4*)(out + (size_t)node * CCH + c4) = o;
}

// ---------------- launch ----------------
extern "C" void kernel_launch(void* const* d_in, const int* in_sizes, int n_in,
                              void* d_out, int out_size, void* d_ws, size_t ws_size,
                              hipStream_t stream) {
    const float* x      = (const float*)d_in[0];   // [N, 128]
    const int*   batch  = (const int*)  d_in[1];   // [N] sorted segment ids (x64 off => int32)
    const float* weight = (const float*)d_in[2];   // [128]
    const float* bias   = (const float*)d_in[3];   // [128]
    float*       out    = (float*)d_out;

    const int N = in_sizes[1];

    // workspace layout
    char*  ws     = (char*)d_ws;
    int*   counts = (int*)ws;                          // 512 ints
    int*   starts = counts + NUM_GRAPHS;               // 512 ints
    float* mean   = (float*)(ws + 4096);               // 512*128 f32
    float* invstd = mean + NUM_GRAPHS * CCH;           // 512*128 f32

    gn_zero_counts<<<1, NUM_GRAPHS, 0, stream>>>(counts);
    gn_count<<<(N + 255) / 256, 256, 0, stream>>>(batch, counts, N);
    gn_scan<<<1, NUM_GRAPHS, 0, stream>>>(counts, starts);
    gn_stats<<<NUM_GRAPHS, 256, 0, stream>>>(x, counts, starts, mean, invstd);

    const int total4 = N * (CCH / 4);
    gn_norm<<<(total4 + 255) / 256, 256, 0, stream>>>(x, batch, weight, bias,
                                                      mean, invstd, out, total4);
}